// ObjectRelationNetwork_34540126994910
// MI455X (gfx1250) — compile-verified
//
#include <hip/hip_runtime.h>

// ---------------------------------------------------------------------------
// ObjectRelationNetwork for MI455X (gfx1250, wave32, WMMA).
//   B=32, T=12, K=16, D=512, H0=512, H1=256.  Slabs = B*(T-1) = 352.
//
//   Factorization: pairs@W0 = O[i]@W0_top + O[j]@W0_bot
//     Kernel A: P' = O@W0_top + b0, Q = O@W0_bot   (per-slab 16x512 tiles)
//               + clipped row sums for is_obj
//     Kernel B: is_obj = clip((t+1)*s_i*s_j, 0, 1)
//     Kernel C: h = relu(P'[i]+Q[j]) (64x512 chunk in LDS),
//               e = relu(h @ W1 + b1) via V_WMMA_F32_16X16X4_F32,
//               W1 staged in LDS k-panels (pair-interleaved -> ds_load_b64,
//               conflict-free, 4x less L2 traffic than per-wave streaming).
// ---------------------------------------------------------------------------

typedef __attribute__((ext_vector_type(2))) float v2f;
typedef __attribute__((ext_vector_type(8))) float v8f;

#define DIMD   512
#define H0D    512
#define H1D    256
#define TM1    11
#define NSLAB  352                 // 32 * 11
#define LDA    516                 // h/A LDS row stride (dwords): 516%64==4 -> conflict-free
#define PSTR   544                 // W1 panel pair-row stride (dwords): 544%64==32
#define HB     0                   // LDS offset of h chunk (dwords)
#define PB     (64 * LDA)          // LDS offset of W1 panel (dwords) = 33024

// ------------------------------ Kernel A -----------------------------------
// One workgroup per slab. LDS: 16 x LDA object tile + 16 rowsum accumulators.
// 8 waves: wave w -> half = w>>2 (0:P via W0 rows [0,512), 1:Q via rows [512,1024)),
//          n0 = (w&3)*128 (8 n-tiles of 16). Accum = 8 x v8f. Waves read
//          disjoint W0 slices -> no intra-WG redundancy, stream from L2.
__global__ void __launch_bounds__(256, 2)
ornet_pq_kernel(const float* __restrict__ O, const float* __restrict__ W0,
                const float* __restrict__ b0, float* __restrict__ Pws,
                float* __restrict__ Qws, float* __restrict__ Sws)
{
    extern __shared__ float lds[];                 // 16*LDA + 16 floats
    const int slab = blockIdx.x;                   // 0..351
    const int b    = slab / TM1;
    const int t    = slab % TM1;
    const int tid  = threadIdx.x;

    // Stage O_t[b,t] (16 x 512, contiguous rows) into LDS with stride LDA.
    const float* Abase = O + ((size_t)(b * 12 + t + 1) * 16) * DIMD;
    {
        const int row = tid >> 4;                  // 0..15
        const int l16 = tid & 15;
        const float4* src = (const float4*)(Abase + (size_t)row * DIMD);
        #pragma unroll
        for (int s = 0; s < 8; ++s) {
            const int c4 = l16 + s * 16;           // float4 column index 0..127
            float4 v = src[c4];
            *(float4*)&lds[row * LDA + c4 * 4] = v;   // row base 2064B -> 16B aligned
        }
    }
    if (tid < 16) lds[16 * LDA + tid] = 0.0f;      // rowsum accumulators
    __syncthreads();

    const int w    = tid >> 5;                     // wave id (wave32)
    const int l    = tid & 31;
    const int half = w >> 2;                       // 0 -> P, 1 -> Q
    const int n0   = (w & 3) * 128;
    const int m    = l & 15;                       // A-frag row
    const int kk   = 2 * (l >> 4);                 // A/B-frag K sub-offset
    const int nl   = l & 15;                       // B/D column lane

    v8f acc[8];
    #pragma unroll
    for (int i = 0; i < 8; ++i) acc[i] = (v8f)0.0f;

    const float* Wh = W0 + (size_t)half * 512 * H0D;   // 512 x 512 half, row-major
    #pragma unroll 2
    for (int k0 = 0; k0 < DIMD; k0 += 4) {
        const v2f a = *(const v2f*)&lds[m * LDA + k0 + kk];   // 8B aligned
        const float* Brow = Wh + (size_t)(k0 + kk) * H0D;
        #pragma unroll
        for (int nt = 0; nt < 8; ++nt) {
            const int n = n0 + nt * 16 + nl;
            v2f bf;
            bf.x = Brow[n];
            bf.y = Brow[H0D + n];
            acc[nt] = __builtin_amdgcn_wmma_f32_16x16x4_f32(
                false, a, false, bf, (short)0, acc[nt], false, false);
        }
    }

    // Store P' = P + b0 (bias folded here), Q raw.
    float* Dst = (half == 0 ? Pws : Qws) + (size_t)slab * 16 * H0D;
    #pragma unroll
    for (int nt = 0; nt < 8; ++nt) {
        const int n = n0 + nt * 16 + nl;
        const float bias = (half == 0) ? b0[n] : 0.0f;
        #pragma unroll
        for (int r = 0; r < 8; ++r) {
            const int mr = r + 8 * (l >> 4);
            Dst[mr * H0D + n] = acc[nt][r] + bias;
        }
    }

    // Clipped per-object row sums (object tile still resident in LDS).
    {
        const int row = tid >> 4, seg = tid & 15;
        float p = 0.0f;
        #pragma unroll
        for (int q = 0; q < 32; ++q) p += lds[row * LDA + seg * 32 + q];
        atomicAdd(&lds[16 * LDA + row], p);        // ds_add_f32
        __syncthreads();
        if (tid < 16) {
            float s = lds[16 * LDA + tid];
            Sws[slab * 16 + tid] = fminf(fmaxf(s, 0.0f), 1.0f);
        }
    }
}

// ------------------------------ Kernel B -----------------------------------
__global__ void __launch_bounds__(256)
ornet_isobj_kernel(const float* __restrict__ Sws, float* __restrict__ isobj)
{
    const int slab = blockIdx.x;
    const int t    = slab % TM1;
    const int tid  = threadIdx.x;                  // p = i*16 + j
    const float si = Sws[slab * 16 + (tid >> 4)];
    const float sj = Sws[slab * 16 + (tid & 15)];
    const float v  = (float)(t + 1) * si * sj;
    isobj[(size_t)slab * 256 + tid] = fminf(fmaxf(v, 0.0f), 1.0f);
}

// ------------------------------ Kernel C -----------------------------------
// Grid = 352*4; each WG handles 64 pair-rows of one slab.
// LDS: h chunk 64 x LDA (132 KB) + W1 k-panel 32 x PSTR (69.6 KB) ~= 197 KB.
// 8 waves: m0=(w&3)*16, n0=(w>>2)*128 -> 8 accum tiles/wave.
// W1 is staged per 64-row k-panel, pair-interleaved so each B fragment is one
// aligned ds_load_b64; all 8 waves share the panel (4x L2 traffic reduction).
__global__ void __launch_bounds__(256, 1)
ornet_e_kernel(const float* __restrict__ Pws, const float* __restrict__ Qws,
               const float* __restrict__ W1, const float* __restrict__ b1,
               float* __restrict__ e_out)
{
    extern __shared__ float lds[];                 // 64*LDA + 32*PSTR floats
    const int slab  = blockIdx.x >> 2;
    const int chunk = blockIdx.x & 3;
    const int tid   = threadIdx.x;

    // Build h[r, :] = relu(P'[i] + Q[j]),  r = 0..63, i = chunk*4 + r/16, j = r%16.
    {
        const int r    = tid >> 2;                 // row 0..63
        const int part = tid & 3;                  // quarter of the 512 columns
        const int i    = chunk * 4 + (r >> 4);
        const int j    = r & 15;
        const float4* Pp = (const float4*)(Pws + (size_t)slab * (16 * H0D) + (size_t)i * H0D);
        const float4* Qp = (const float4*)(Qws + (size_t)slab * (16 * H0D) + (size_t)j * H0D);
        #pragma unroll
        for (int s = 0; s < 32; ++s) {
            const int c4 = part * 32 + s;          // float4 col 0..127
            float4 p = Pp[c4], q = Qp[c4], h;
            h.x = fmaxf(p.x + q.x, 0.0f);
            h.y = fmaxf(p.y + q.y, 0.0f);
            h.z = fmaxf(p.z + q.z, 0.0f);
            h.w = fmaxf(p.w + q.w, 0.0f);
            *(float4*)&lds[HB + r * LDA + c4 * 4] = h;
        }
    }

    const int w    = tid >> 5;
    const int l    = tid & 31;
    const int m0   = (w & 3) * 16;
    const int n0   = (w >> 2) * 128;
    const int mlds = m0 + (l & 15);
    const int kk   = 2 * (l >> 4);
    const int nl   = l & 15;

    v8f acc[8];
    #pragma unroll
    for (int i = 0; i < 8; ++i) acc[i] = (v8f)0.0f;

    for (int kp = 0; kp < 8; ++kp) {
        __syncthreads();   // (first iter: h ready; later: waves done with panel)

        // Stage 64 k-rows of W1 into pair-interleaved panel:
        //   element (k,n) -> lds[PB + (k>>1)*PSTR + 2n + (k&1)]
        {
            const float4* Wp = (const float4*)(W1 + (size_t)kp * 64 * H1D);
            #pragma unroll
            for (int s = 0; s < 16; ++s) {
                const int f4 = s * 256 + tid;      // 0..4095
                const int k  = f4 >> 6;            // 0..63 (panel-local)
                const int n  = (f4 & 63) * 4;
                float4 v = Wp[f4];
                float* dst = &lds[PB + (k >> 1) * PSTR + (k & 1)];
                dst[(n + 0) * 2] = v.x;
                dst[(n + 1) * 2] = v.y;
                dst[(n + 2) * 2] = v.z;
                dst[(n + 3) * 2] = v.w;
            }
        }
        __syncthreads();

        #pragma unroll 2
        for (int kq = 0; kq < 64; kq += 4) {
            const int k0 = kp * 64 + kq;
            const v2f a = *(const v2f*)&lds[HB + mlds * LDA + k0 + kk];
            const int pr = (kq + kk) >> 1;         // panel pair-row (even k)
            #pragma unroll
            for (int nt = 0; nt < 8; ++nt) {
                const int n = n0 + nt * 16 + nl;
                const v2f bf = *(const v2f*)&lds[PB + pr * PSTR + n * 2]; // ds_load_b64
                acc[nt] = __builtin_amdgcn_wmma_f32_16x16x4_f32(
                    false, a, false, bf, (short)0, acc[nt], false, false);
            }
        }
    }

    // e = relu(acc + b1), e[(slab*256 + chunk*64 + m), n]
    float* Obase = e_out + ((size_t)slab * 256 + chunk * 64 + m0) * H1D;
    #pragma unroll
    for (int nt = 0; nt < 8; ++nt) {
        const int n = n0 + nt * 16 + nl;
        const float bias = b1[n];
        #pragma unroll
        for (int r = 0; r < 8; ++r) {
            const int mr = r + 8 * (l >> 4);
            Obase[(size_t)mr * H1D + n] = fmaxf(acc[nt][r] + bias, 0.0f);
        }
    }
}

// ------------------------------ Launch -------------------------------------
extern "C" void kernel_launch(void* const* d_in, const int* in_sizes, int n_in,
                              void* d_out, int out_size, void* d_ws, size_t ws_size,
                              hipStream_t stream)
{
    (void)in_sizes; (void)n_in; (void)out_size; (void)ws_size;
    const float* O  = (const float*)d_in[0];   // (32,12,16,512)
    const float* W0 = (const float*)d_in[1];   // (1024,512)
    const float* b0 = (const float*)d_in[2];   // (512,)
    const float* W1 = (const float*)d_in[3];   // (512,256)
    const float* b1 = (const float*)d_in[4];   // (256,)
    // d_in[5] = D (constant 512), unused.

    float* out      = (float*)d_out;
    float* e_out    = out;                                   // 32*11*256*256
    float* isobj    = out + (size_t)NSLAB * 256 * H1D;       // + 32*11*256

    float* Pws = (float*)d_ws;                               // 352*16*512
    float* Qws = Pws + (size_t)NSLAB * 16 * H0D;             // 352*16*512
    float* Sws = Qws + (size_t)NSLAB * 16 * H0D;             // 352*16

    ornet_pq_kernel<<<NSLAB, 256, (16 * LDA + 16) * sizeof(float), stream>>>(
        O, W0, b0, Pws, Qws, Sws);
    ornet_isobj_kernel<<<NSLAB, 256, 0, stream>>>(Sws, isobj);
    ornet_e_kernel<<<NSLAB * 4, 256, (64 * LDA + 32 * PSTR) * sizeof(float), stream>>>(
        Pws, Qws, W1, b1, e_out);
}